// GNN_6365141532848
// MI455X (gfx1250) — compile-verified
//
#include <hip/hip_runtime.h>

typedef __attribute__((ext_vector_type(2))) float v2f;
typedef __attribute__((ext_vector_type(8))) float v8f;

#define D 128  // D_IN == D_OUT == 128

// CDNA5 async global->LDS copy path (ASYNCcnt), if this toolchain exposes it.
#if defined(__has_builtin)
#if __has_builtin(__builtin_amdgcn_global_load_async_to_lds_b128)
#define GCN_ASYNC_LDS 1
#endif
#endif

// Exact parameter types per hipcc diagnostic:
//   arg0: int __attribute__((vector_size(16))) addrspace(1)*   (global, non-const)
//   arg1: matching 16-byte vector in LDS addrspace(3)
typedef int b128_t __attribute__((vector_size(4 * sizeof(int))));
typedef __attribute__((address_space(1))) b128_t gb128_t;
typedef __attribute__((address_space(3))) b128_t lb128_t;

// -------------------------------------------------------------------------
// h = x @ W^T via V_WMMA_F32_16X16X4_F32 (full f32 precision, wave32 WMMA).
// One block = 16 node rows; 8 waves, one 16-wide output column tile each.
// x tile (16x128 f32, 8KB) staged in LDS once (async DMA when available);
// W (64KB) is L2-resident.
// -------------------------------------------------------------------------
__global__ void __launch_bounds__(256) gcn_gemm_wmma(
    const float* __restrict__ x, const float* __restrict__ W,
    float* __restrict__ h)
{
    __shared__ float xs[16 * D];  // 8 KB x-tile

    const int row0 = blockIdx.x * 16;
    const int tid  = threadIdx.x;

    // Cooperative stage of 16x128 f32 tile (2048 floats = 512 float4).
    const float4* xg  = reinterpret_cast<const float4*>(x + (size_t)row0 * D);
    float4*       xs4 = reinterpret_cast<float4*>(xs);

#ifdef GCN_ASYNC_LDS
    // Direct memory->LDS DMA: global_load_async_to_lds_b128 (ASYNCcnt).
    __builtin_amdgcn_global_load_async_to_lds_b128(
        (gb128_t*)(unsigned long long)(xg + tid),
        (lb128_t*)(unsigned)(unsigned long long)(xs4 + tid), 0, 0);
    __builtin_amdgcn_global_load_async_to_lds_b128(
        (gb128_t*)(unsigned long long)(xg + tid + 256),
        (lb128_t*)(unsigned)(unsigned long long)(xs4 + tid + 256), 0, 0);
#if __has_builtin(__builtin_amdgcn_s_wait_asynccnt)
    __builtin_amdgcn_s_wait_asynccnt(0);
#else
    asm volatile("s_wait_asynccnt 0x0" ::: "memory");
#endif
#else
    xs4[tid]       = xg[tid];
    xs4[tid + 256] = xg[tid + 256];
#endif
    __syncthreads();

    const int wave = tid >> 5;
    const int lane = tid & 31;
    const int col0 = wave * 16;
    const int m    = lane & 15;        // A row / B col within tile
    const int kh   = (lane >> 4) << 1; // lanes 16-31 hold K+2,K+3 (ISA layout)

    // B[k][n] = W[col0+n][k]  (B = W^T tile)
    const float* wrow = W + (size_t)(col0 + m) * D;

    v8f c = {};
    for (int k0 = 0; k0 < D; k0 += 4) {
        v2f a = *reinterpret_cast<const v2f*>(&xs[m * D + k0 + kh]);
        v2f b = *reinterpret_cast<const v2f*>(&wrow[k0 + kh]);
        // (neg_a, A, neg_b, B, c_mod, C, reuse_a, reuse_b)
        c = __builtin_amdgcn_wmma_f32_16x16x4_f32(
                false, a, false, b, (short)0, c, false, false);
    }

    // C/D layout: VGPR v -> M=v (lanes 0-15), M=v+8 (lanes 16-31); N = lane%16
    const int n  = lane & 15;
    const int mb = (lane >> 4) << 3;  // 0 or 8
    float* hout = h + (size_t)(row0 + mb) * D + col0 + n;
#pragma unroll
    for (int v = 0; v < 8; ++v)
        hout[(size_t)v * D] = c[v];
}

// deg[n] = 1.0f  (self loop)
__global__ void gcn_deg_init(float* __restrict__ deg, int N)
{
    int i = blockIdx.x * blockDim.x + threadIdx.x;
    if (i < N) deg[i] = 1.0f;
}

// deg[dst[e]] += 1
__global__ void gcn_deg_count(const int* __restrict__ dst,
                              float* __restrict__ deg, int E)
{
    int e = blockIdx.x * blockDim.x + threadIdx.x;
    if (e < E) atomicAdd(&deg[dst[e]], 1.0f);
}

// dis[n] = rsqrt(deg[n])   (deg >= 1 always, matches reference's guard)
__global__ void gcn_rsqrt(float* __restrict__ deg, int N)
{
    int i = blockIdx.x * blockDim.x + threadIdx.x;
    if (i < N) deg[i] = rsqrtf(deg[i]);
}

// out[n][d] = h[n][d] * dis[n]^2   (self-loop message; also inits out)
__global__ void gcn_selfloop(const float* __restrict__ h,
                             const float* __restrict__ dis,
                             float* __restrict__ out, int total)
{
    int i = blockIdx.x * blockDim.x + threadIdx.x;
    if (i < total) {
        float s = dis[i >> 7];
        out[i] = h[i] * (s * s);
    }
}

// One wave per edge: gather h[src] (float4/lane, 512B coalesced),
// scale by dis[src]*dis[dst], scatter via global_atomic_add_f32.
__global__ void __launch_bounds__(256) gcn_scatter(
    const int* __restrict__ src, const int* __restrict__ dst,
    const float* __restrict__ h, const float* __restrict__ dis,
    float* __restrict__ out, int E)
{
    const int e = blockIdx.x * (blockDim.x >> 5) + (threadIdx.x >> 5);
    if (e >= E) return;
    const int lane = threadIdx.x & 31;

    const int s = src[e];
    const int d = dst[e];
    const float norm = dis[s] * dis[d];

    const float4 m4 = reinterpret_cast<const float4*>(h + (size_t)s * D)[lane];
    float* od = out + (size_t)d * D + lane * 4;
    atomicAdd(od + 0, m4.x * norm);
    atomicAdd(od + 1, m4.y * norm);
    atomicAdd(od + 2, m4.z * norm);
    atomicAdd(od + 3, m4.w * norm);
}

// out = relu(out + b)
__global__ void gcn_bias_relu(float* __restrict__ out,
                              const float* __restrict__ b, int total)
{
    int i = blockIdx.x * blockDim.x + threadIdx.x;
    if (i < total) out[i] = fmaxf(out[i] + b[i & (D - 1)], 0.0f);
}

extern "C" void kernel_launch(void* const* d_in, const int* in_sizes, int n_in,
                              void* d_out, int out_size, void* d_ws, size_t ws_size,
                              hipStream_t stream)
{
    const float* x  = (const float*)d_in[0];
    const int*   ei = (const int*)d_in[1];   // [2, E]: row0 = src, row1 = dst
    const float* W  = (const float*)d_in[2];
    const float* b  = (const float*)d_in[3];

    const int N = in_sizes[0] / D;   // 100000
    const int E = in_sizes[1] / 2;   // 1600000
    const int* src = ei;
    const int* dst = ei + E;

    // Workspace layout: h [N*D] f32, then dis [N] f32 (~51.6 MB total)
    float* h   = (float*)d_ws;
    float* dis = h + (size_t)N * D;

    float* out = (float*)d_out;
    const int total = N * D;

    // 1) h = x @ W^T  (WMMA f32)
    gcn_gemm_wmma<<<N / 16, 256, 0, stream>>>(x, W, h);

    // 2) degrees (incl. self loop) -> dis = rsqrt(deg)
    gcn_deg_init <<<(N + 255) / 256, 256, 0, stream>>>(dis, N);
    gcn_deg_count<<<(E + 255) / 256, 256, 0, stream>>>(dst, dis, E);
    gcn_rsqrt    <<<(N + 255) / 256, 256, 0, stream>>>(dis, N);

    // 3) out = h * dis^2 (self loop, initializes d_out)
    gcn_selfloop<<<(total + 255) / 256, 256, 0, stream>>>(h, dis, out, total);

    // 4) edge messages: wave-per-edge gather/scale/atomic-scatter
    gcn_scatter<<<(E + 7) / 8, 256, 0, stream>>>(src, dst, h, dis, out, E);

    // 5) bias + ReLU
    gcn_bias_relu<<<(total + 255) / 256, 256, 0, stream>>>(out, b, total);
}